// GCN_88021059764774
// MI455X (gfx1250) — compile-verified
//
#include <hip/hip_runtime.h>
#include <hip/hip_bf16.h>

// ---------------------------------------------------------------------------
// GCN on MI455X (gfx1250).
//  - GEMMs use V_WMMA_F32_16X16X4_F32 (full fp32, wave32, 16x16 tiles).
//  - A-tile staged into LDS via GLOBAL_LOAD_ASYNC_TO_LDS_B128 (ASYNCcnt path)
//    when the toolchain exposes the builtin; cooperative fill otherwise.
//  - Edge scatter uses wave-per-edge float4 gather + global_atomic_add_f32;
//    h (25.6 MB) lives in the 192 MB L2, so scatter traffic stays on-chip.
// ---------------------------------------------------------------------------

typedef float v2f __attribute__((ext_vector_type(2)));
typedef float v8f __attribute__((ext_vector_type(8)));
typedef int   v4i __attribute__((ext_vector_type(4)));

typedef __attribute__((address_space(1))) v4i* gbl_v4i_p;   // global
typedef __attribute__((address_space(3))) v4i* lds_v4i_p;   // LDS

#define FEAT_IN  64
#define FEAT_H   128
#define NCLASS   10
#define NGRAPH   64

#if defined(__has_builtin)
#if __has_builtin(__builtin_amdgcn_global_load_async_to_lds_b128)
#define HAVE_ASYNC_LDS 1
#endif
#if __has_builtin(__builtin_amdgcn_s_wait_asynccnt)
#define HAVE_WAIT_ASYNC 1
#endif
#endif

// ------------------------------- utility -----------------------------------
__global__ void k_zero(float* __restrict__ p, int n) {
    int i = blockIdx.x * blockDim.x + threadIdx.x;
    if (i < n) p[i] = 0.0f;
}

// deg accumulation over dst (real edges only; self-loop folded into finalize)
__global__ void k_degree(const int* __restrict__ dst, float* __restrict__ deg, int nE) {
    int i = blockIdx.x * blockDim.x + threadIdx.x;
    if (i < nE) unsafeAtomicAdd(&deg[dst[i]], 1.0f);
}

// dis[i] = rsqrt(deg[i] + 1)   (self-loop guarantees deg > 0)
__global__ void k_deg_finalize(float* __restrict__ deg, int n) {
    int i = blockIdx.x * blockDim.x + threadIdx.x;
    if (i < n) deg[i] = rsqrtf(deg[i] + 1.0f);
}

// ------------------------------ WMMA GEMM ----------------------------------
// out[M,128] = A[M,K] @ W[K,128].  Block = 256 threads (8 waves), grid = M/16.
// Each wave computes one 16x16 tile of the 16x128 block-row using
// v_wmma_f32_16x16x4_f32.  A-tile staged in LDS (stride K+4: rows 16B-aligned
// for async B128 writes, banks spread 4*m+k mod 64 -> conflict-free).
//
// Fragment layouts (ISA 7.12.2, 32-bit, wave32):
//   A 16x4 : lane group g = lane>>4, m = lane&15 ; VGPR v holds K = k0+2g+v
//   B 4x16 : (symmetric assumption) VGPR v holds K = k0+2g+v, N = n0+(lane&15)
//   C 16x16: VGPR r -> row r + 8g, col n0+(lane&15)
template <int K>
__global__ __launch_bounds__(256) void k_gemm_wmma(const float* __restrict__ A,
                                                   const float* __restrict__ W,
                                                   float* __restrict__ out, int M) {
    __shared__ float As[16][K + 4];
    const int m0 = blockIdx.x * 16;

#if defined(HAVE_ASYNC_LDS)
    // Async DMA: 16 rows x (K/4) 16-byte chunks, bypassing VGPRs (ASYNCcnt).
    constexpr int NCH = 16 * (K / 4);
    for (int c = threadIdx.x; c < NCH; c += 256) {
        const int r  = c / (K / 4);
        const int kc = (c - r * (K / 4)) * 4;
        __builtin_amdgcn_global_load_async_to_lds_b128(
            (gbl_v4i_p)(A + (size_t)(m0 + r) * K + kc),
            (lds_v4i_p)&As[r][kc],
            0, 0);
    }
#if defined(HAVE_WAIT_ASYNC)
    __builtin_amdgcn_s_wait_asynccnt(0);
#else
    asm volatile("s_wait_asynccnt 0x0" ::: "memory");
#endif
#else
    for (int idx = threadIdx.x; idx < 16 * K; idx += 256) {
        int r = idx / K, k = idx - r * K;
        As[r][k] = A[(size_t)(m0 + r) * K + k];
    }
#endif
    __syncthreads();

    const int wave = threadIdx.x >> 5;          // 0..7 -> output column tile
    const int lane = threadIdx.x & 31;
    const int g    = lane >> 4;                 // half-wave group
    const int mrow = lane & 15;
    const int n    = wave * 16 + (lane & 15);

    v8f c = {0.f, 0.f, 0.f, 0.f, 0.f, 0.f, 0.f, 0.f};
#pragma unroll
    for (int k0 = 0; k0 < K; k0 += 4) {
        const int ka = k0 + 2 * g;
        v2f a, b;
        a.x = As[mrow][ka];
        a.y = As[mrow][ka + 1];
        b.x = W[ka * FEAT_H + n];
        b.y = W[(ka + 1) * FEAT_H + n];
        c = __builtin_amdgcn_wmma_f32_16x16x4_f32(false, a, false, b,
                                                  (short)0, c, false, false);
    }

#pragma unroll
    for (int r = 0; r < 8; ++r) {
        const int row = m0 + r + 8 * g;
        out[(size_t)row * FEAT_H + n] = c[r];
    }
}

// ------------------------------ edge scatter -------------------------------
// One wave (32 lanes) per edge; each lane handles 4 of the 128 features.
// agg[dst] += dis[src]*dis[dst] * h[src]
__global__ void k_scatter(const float* __restrict__ h, const int* __restrict__ src,
                          const int* __restrict__ dst, const float* __restrict__ dis,
                          float* __restrict__ agg, int nE) {
    const int edge = blockIdx.x * (blockDim.x >> 5) + (threadIdx.x >> 5);
    if (edge >= nE) return;
    const int lane = threadIdx.x & 31;
    const int s = src[edge], d = dst[edge];
    const float nrm = dis[s] * dis[d];
    const float4 v = reinterpret_cast<const float4*>(h + (size_t)s * FEAT_H)[lane];
    float* o = agg + (size_t)d * FEAT_H + lane * 4;
    unsafeAtomicAdd(o + 0, v.x * nrm);
    unsafeAtomicAdd(o + 1, v.y * nrm);
    unsafeAtomicAdd(o + 2, v.z * nrm);
    unsafeAtomicAdd(o + 3, v.w * nrm);
}

// epilogue: agg = relu(agg + dis[i]^2 * h[i] + b)   (self-loop + bias + relu)
__global__ void k_selfloop_bias_relu(const float* __restrict__ h,
                                     const float* __restrict__ dis,
                                     const float* __restrict__ b,
                                     float* __restrict__ agg, int total) {
    int idx = blockIdx.x * blockDim.x + threadIdx.x;
    if (idx >= total) return;
    const int i = idx >> 7;          // /128
    const int c = idx & 127;
    const float ds = dis[i];
    const float v = agg[idx] + ds * ds * h[idx] + b[c];
    agg[idx] = v > 0.0f ? v : 0.0f;
}

// ------------------------------- pooling -----------------------------------
// wave-per-node: pooled[batch[i]] += a[i]; counts[batch[i]] += 1
__global__ void k_pool(const float* __restrict__ a, const int* __restrict__ batch,
                       float* __restrict__ pooled, float* __restrict__ counts, int n) {
    const int node = blockIdx.x * (blockDim.x >> 5) + (threadIdx.x >> 5);
    if (node >= n) return;
    const int lane = threadIdx.x & 31;
    const int gidx = batch[node];
    const float4 v = reinterpret_cast<const float4*>(a + (size_t)node * FEAT_H)[lane];
    float* o = pooled + (size_t)gidx * FEAT_H + lane * 4;
    unsafeAtomicAdd(o + 0, v.x);
    unsafeAtomicAdd(o + 1, v.y);
    unsafeAtomicAdd(o + 2, v.z);
    unsafeAtomicAdd(o + 3, v.w);
    if (lane == 0) unsafeAtomicAdd(&counts[gidx], 1.0f);
}

// logits[g,j] = dot(pooled_sum[g], Wl[:,j]) / max(count,1) + bl[j]
__global__ void k_logits(const float* __restrict__ pooled, const float* __restrict__ counts,
                         const float* __restrict__ Wl, const float* __restrict__ bl,
                         float* __restrict__ out) {
    const int g = blockIdx.x;
    const int j = threadIdx.x;
    if (j >= NCLASS) return;
    float cnt = counts[g];
    cnt = cnt > 1.0f ? cnt : 1.0f;
    float s = 0.0f;
    const float* pr = pooled + g * FEAT_H;
#pragma unroll 8
    for (int c = 0; c < FEAT_H; ++c) s += pr[c] * Wl[c * NCLASS + j];
    out[g * NCLASS + j] = s / cnt + bl[j];
}

// ------------------------------- launcher ----------------------------------
extern "C" void kernel_launch(void* const* d_in, const int* in_sizes, int n_in,
                              void* d_out, int out_size, void* d_ws, size_t ws_size,
                              hipStream_t stream) {
    const float* x     = (const float*)d_in[0];
    const int*   edges = (const int*)d_in[1];
    const int*   batch = (const int*)d_in[2];
    const float* W1    = (const float*)d_in[3];
    const float* b1    = (const float*)d_in[4];
    const float* W2    = (const float*)d_in[5];
    const float* b2    = (const float*)d_in[6];
    const float* Wl    = (const float*)d_in[7];
    const float* bl    = (const float*)d_in[8];
    float* out = (float*)d_out;

    const int N  = in_sizes[0] / FEAT_IN;   // 50000
    const int E  = in_sizes[1] / 2;         // 800000
    const int* src = edges;
    const int* dst = edges + E;

    // workspace layout (fp32), 256-float aligned chunks
    float* ws = (float*)d_ws;
    size_t o = 0;
    float* dis    = ws + o;  o += ((size_t)N + 255) & ~(size_t)255;
    float* bufA   = ws + o;  o += (size_t)N * FEAT_H;          // h (GEMM out)
    float* bufB   = ws + o;  o += (size_t)N * FEAT_H;          // agg / activation
    float* pooled = ws + o;  o += NGRAPH * FEAT_H;
    float* counts = ws + o;  o += NGRAPH;

    const int NH = N * FEAT_H;
    const int TB = 256;
    auto cdiv = [](int a, int b) { return (a + b - 1) / b; };

    // degrees -> dis = rsqrt(deg+1)
    k_zero<<<cdiv(N, TB), TB, 0, stream>>>(dis, N);
    k_degree<<<cdiv(E, TB), TB, 0, stream>>>(dst, dis, E);
    k_deg_finalize<<<cdiv(N, TB), TB, 0, stream>>>(dis, N);

    // ---- layer 1 ----
    k_gemm_wmma<FEAT_IN><<<N / 16, TB, 0, stream>>>(x, W1, bufA, N);
    k_zero<<<cdiv(NH, TB), TB, 0, stream>>>(bufB, NH);
    k_scatter<<<cdiv(E, TB / 32), TB, 0, stream>>>(bufA, src, dst, dis, bufB, E);
    k_selfloop_bias_relu<<<cdiv(NH, TB), TB, 0, stream>>>(bufA, dis, b1, bufB, NH);

    // ---- layer 2 ----
    k_gemm_wmma<FEAT_H><<<N / 16, TB, 0, stream>>>(bufB, W2, bufA, N);
    k_zero<<<cdiv(NH, TB), TB, 0, stream>>>(bufB, NH);
    k_scatter<<<cdiv(E, TB / 32), TB, 0, stream>>>(bufA, src, dst, dis, bufB, E);
    k_selfloop_bias_relu<<<cdiv(NH, TB), TB, 0, stream>>>(bufA, dis, b2, bufB, NH);

    // ---- pool + head ----
    k_zero<<<cdiv(NGRAPH * FEAT_H + NGRAPH, TB), TB, 0, stream>>>(pooled, NGRAPH * FEAT_H + NGRAPH);
    k_pool<<<cdiv(N, TB / 32), TB, 0, stream>>>(bufB, batch, pooled, counts, N);
    k_logits<<<NGRAPH, 32, 0, stream>>>(pooled, counts, Wl, bl, out);
}